// Keap1PredictorGNN_88708254531587
// MI455X (gfx1250) — compile-verified
//
#include <hip/hip_runtime.h>
#include <hip/hip_bf16.h>

// ---------------------------------------------------------------------------
// Keap1PredictorGNN for MI455X (gfx1250, wave32).
//   GAT layer = GEMM + per-node head projections + 3 bandwidth-bound edge
//   sweeps (atomic max-key / exp-sum / weighted scatter), then global mean
//   pool and two KAN layers.
//   Layer-2 GEMM (50000x256 @ 256x256, fp32): 8-wave blocks compute 128x64
//   tiles; the B panel (256x64 = 64KB) is staged in LDS with CDNA5 async
//   GLOBAL_LOAD_ASYNC_TO_LDS_B128, and each wave reuses one A fragment across
//   4 V_WMMA_F32_16X16X4_F32 issues (exact fp32 semantics).
// ---------------------------------------------------------------------------

#define HEADS 4
#define HIDC  64
#define D1    256          // HEADS*HIDC

typedef __attribute__((ext_vector_type(2))) float v2f;
typedef __attribute__((ext_vector_type(8))) float v8f;

// ---- monotone float<->uint key for native atomic max on floats -------------
__device__ __forceinline__ unsigned fkey(float x) {
  unsigned b = __float_as_uint(x);
  return (b & 0x80000000u) ? ~b : (b | 0x80000000u);
}
__device__ __forceinline__ float funkey(unsigned k) {
  return __uint_as_float((k & 0x80000000u) ? (k & 0x7fffffffu) : ~k);
}

__device__ __forceinline__ float leaky(float x) { return x > 0.f ? x : 0.2f * x; }
__device__ __forceinline__ float elu1(float x)  { return x > 0.f ? x : (__expf(x) - 1.f); }
__device__ __forceinline__ float silu(float x)  { return x / (1.f + __expf(-x)); }

// Cubic B-spline basis on uniform knots g_j = 0.4*j - 2.2 (12 knots -> 8 basis)
__device__ __forceinline__ void bspline8(float x, float b[8]) {
  float bb[11];
#pragma unroll
  for (int j = 0; j < 11; ++j) {
    float g0 = 0.4f * j - 2.2f;
    float g1 = g0 + 0.4f;
    bb[j] = (x >= g0 && x < g1) ? 1.f : 0.f;
  }
#pragma unroll
  for (int k = 1; k <= 3; ++k) {
    float inv = 1.f / (0.4f * (float)k);
#pragma unroll
    for (int j = 0; j < 11 - k; ++j) {
      float gj   = 0.4f * j - 2.2f;
      float gjk1 = 0.4f * (j + k + 1) - 2.2f;
      bb[j] = (x - gj) * inv * bb[j] + (gjk1 - x) * inv * bb[j + 1];
    }
  }
#pragma unroll
  for (int p = 0; p < 8; ++p) b[p] = bb[p];
}

// ---------------------------------------------------------------------------
// Layer-1 GEMM: xw[n][j] = sum_k x[n][k] * W1[k][j]   (K=6, bandwidth bound)
// ---------------------------------------------------------------------------
__global__ __launch_bounds__(256) void k_gemm_small(const float* __restrict__ x,
                                                    const float* __restrict__ W,
                                                    float* __restrict__ xw) {
  int n = blockIdx.x, j = threadIdx.x;
  float acc = 0.f;
#pragma unroll
  for (int k = 0; k < 6; ++k) acc += x[n * 6 + k] * W[k * D1 + j];
  xw[n * D1 + j] = acc;
}

// ---------------------------------------------------------------------------
// Layer-2 GEMM: C[M,256] = A[M,256] @ B[256,256].
// Block = 8 waves -> 128x64 output tile at (blockIdx.x*128, blockIdx.y*64).
// Stage B panel (256 rows x 64 cols, 64KB) in LDS via async-to-LDS b128,
// then each wave computes one 16-row strip: per K-step one A fragment is
// reused across 4 N-subtiles (4 wmma issues, B fragments from LDS).
// Fragment layouts per ISA 7.12.2:
//   A (16x4 f32): lane<16 row=lane K=k0+{0,1}; lane>=16 K=k0+{2,3}
//   B (4x16):     v0 = row k0+2*half, v1 = row k0+1+2*half, col = lane&15
//   C/D (16x16):  vgpr r -> row r + 8*half, col = lane&15
// ---------------------------------------------------------------------------
__global__ __launch_bounds__(256) void k_gemm_wmma(const float* __restrict__ A,
                                                   const float* __restrict__ B,
                                                   float* __restrict__ C, int M) {
  __shared__ float Bs[D1 * 64];          // 64 KB panel: Bs[row][col0..col0+63]
  const int t    = threadIdx.x;
  const int col0 = blockIdx.y * 64;

  // ---- cooperative async stage of B panel: 16384 floats = 4096 b128 -------
  {
    unsigned ldsBase = (unsigned)(uintptr_t)&Bs[0];
#pragma unroll
    for (int i = 0; i < 16; ++i) {
      int f   = (i * 256 + t) * 4;       // flat float index into panel
      int row = f >> 6;
      int col = f & 63;
      unsigned goff  = (unsigned)((row * D1 + col0 + col) * 4);
      unsigned laddr = ldsBase + (unsigned)(f * 4);
      asm volatile("global_load_async_to_lds_b128 %0, %1, %2"
                   :: "v"(laddr), "v"(goff), "s"(B)
                   : "memory");
    }
    asm volatile("s_wait_asynccnt 0x0" ::: "memory");
  }
  __syncthreads();

  const int wave = t >> 5;
  const int lane = t & 31;
  const int half = lane >> 4;
  const int l15  = lane & 15;
  const int mt   = blockIdx.x * 8 + wave;      // 16-row tile index
  if (mt * 16 >= M) return;                    // whole-wave exit after barrier

  const int arow = mt * 16 + l15;
  v8f c0 = {}, c1 = {}, c2 = {}, c3 = {};
  for (int k0 = 0; k0 < D1; k0 += 4) {
    const int ka = k0 + half * 2;
    v2f a;
    a.x = A[arow * D1 + ka];
    a.y = A[arow * D1 + ka + 1];
    v2f b;
    b.x = Bs[ka * 64 + l15];
    b.y = Bs[(ka + 1) * 64 + l15];
    c0 = __builtin_amdgcn_wmma_f32_16x16x4_f32(false, a, false, b, (short)0, c0, false, false);
    b.x = Bs[ka * 64 + 16 + l15];
    b.y = Bs[(ka + 1) * 64 + 16 + l15];
    c1 = __builtin_amdgcn_wmma_f32_16x16x4_f32(false, a, false, b, (short)0, c1, false, false);
    b.x = Bs[ka * 64 + 32 + l15];
    b.y = Bs[(ka + 1) * 64 + 32 + l15];
    c2 = __builtin_amdgcn_wmma_f32_16x16x4_f32(false, a, false, b, (short)0, c2, false, false);
    b.x = Bs[ka * 64 + 48 + l15];
    b.y = Bs[(ka + 1) * 64 + 48 + l15];
    c3 = __builtin_amdgcn_wmma_f32_16x16x4_f32(false, a, false, b, (short)0, c3, false, false);
  }
  const int crow = mt * 16 + half * 8;
#pragma unroll
  for (int r = 0; r < 8; ++r) {
    C[(crow + r) * D1 + col0 + l15]      = c0[r];
    C[(crow + r) * D1 + col0 + 16 + l15] = c1[r];
    C[(crow + r) * D1 + col0 + 32 + l15] = c2[r];
    C[(crow + r) * D1 + col0 + 48 + l15] = c3[r];
  }
}

// ---------------------------------------------------------------------------
// Per-node head projections: alpha_s[n][h] = sum_c xw[n][h*64+c]*a_src[h][c]
// One block per node, 64-thread-group LDS tree reduction per head.
// ---------------------------------------------------------------------------
__global__ __launch_bounds__(256) void k_alpha(const float* __restrict__ xw,
                                               const float* __restrict__ a_src,
                                               const float* __restrict__ a_dst,
                                               float* __restrict__ as,
                                               float* __restrict__ ad) {
  __shared__ float ss[D1];
  __shared__ float sd[D1];
  int n = blockIdx.x, j = threadIdx.x;
  float v = xw[n * D1 + j];
  ss[j] = v * a_src[j];
  sd[j] = v * a_dst[j];
  __syncthreads();
#pragma unroll
  for (int s = 32; s >= 1; s >>= 1) {
    if ((j & 63) < s) { ss[j] += ss[j + s]; sd[j] += sd[j + s]; }
    __syncthreads();
  }
  if ((j & 63) == 0) {
    int h = j >> 6;
    as[n * HEADS + h] = ss[j];
    ad[n * HEADS + h] = sd[j];
  }
}

// ---------------------------------------------------------------------------
__global__ void k_init_nodes(float* __restrict__ acc, unsigned* __restrict__ mkey,
                             float* __restrict__ den, int N) {
  int i = blockIdx.x * blockDim.x + threadIdx.x;
  if (i < N * D1) acc[i] = 0.f;
  if (i < N * HEADS) { mkey[i] = 0u; den[i] = 0.f; }
}

// edge sweep 1: segment max (as monotone-uint atomic max)
__global__ void k_edge_max(const long long* __restrict__ ei, int E, int N,
                           const float* __restrict__ as,
                           const float* __restrict__ ad,
                           unsigned* __restrict__ mkey) {
  int idx = blockIdx.x * blockDim.x + threadIdx.x;
  int tot = E + N;
  if (idx >= tot) return;
  int s, d;
  if (idx < E) { s = (int)ei[idx]; d = (int)ei[E + idx]; }
  else         { s = d = idx - E; }
#pragma unroll
  for (int h = 0; h < HEADS; ++h) {
    float e = leaky(as[s * HEADS + h] + ad[d * HEADS + h]);
    atomicMax(&mkey[d * HEADS + h], fkey(e));
  }
}

// edge sweep 2: denominator
__global__ void k_edge_exp(const long long* __restrict__ ei, int E, int N,
                           const float* __restrict__ as,
                           const float* __restrict__ ad,
                           const unsigned* __restrict__ mkey,
                           float* __restrict__ den) {
  int idx = blockIdx.x * blockDim.x + threadIdx.x;
  int tot = E + N;
  if (idx >= tot) return;
  int s, d;
  if (idx < E) { s = (int)ei[idx]; d = (int)ei[E + idx]; }
  else         { s = d = idx - E; }
#pragma unroll
  for (int h = 0; h < HEADS; ++h) {
    float e = leaky(as[s * HEADS + h] + ad[d * HEADS + h]);
    float m = funkey(mkey[d * HEADS + h]);
    atomicAdd(&den[d * HEADS + h], __expf(e - m));
  }
}

// edge sweep 3: alpha-weighted message scatter (256 lanes per edge, coalesced;
// next edge's source row prefetched to overlap gather with atomic drain)
__global__ __launch_bounds__(256) void k_edge_scatter(const long long* __restrict__ ei,
                                                      int E, int N,
                                                      const float* __restrict__ as,
                                                      const float* __restrict__ ad,
                                                      const unsigned* __restrict__ mkey,
                                                      const float* __restrict__ den,
                                                      const float* __restrict__ xw,
                                                      float* __restrict__ acc) {
  int j = threadIdx.x;
  int h = j >> 6;
  int tot = E + N;
  for (int e = blockIdx.x; e < tot; e += gridDim.x) {
    int s, d;
    if (e < E) { s = (int)ei[e]; d = (int)ei[E + e]; }
    else       { s = d = e - E; }
    int en = e + gridDim.x;                       // prefetch next gather row
    if (en < tot) {
      int sn = (en < E) ? (int)ei[en] : (en - E);
      __builtin_prefetch(&xw[sn * D1 + j], 0, 0);
    }
    float t = leaky(as[s * HEADS + h] + ad[d * HEADS + h]);
    float m = funkey(mkey[d * HEADS + h]);
    float alpha = __expf(t - m) / (den[d * HEADS + h] + 1e-16f);
    atomicAdd(&acc[d * D1 + j], alpha * xw[s * D1 + j]);
  }
}

__global__ void k_bias_elu(const float* __restrict__ acc, const float* __restrict__ bias,
                           float* __restrict__ h, int total) {
  int i = blockIdx.x * blockDim.x + threadIdx.x;
  if (i >= total) return;
  h[i] = elu1(acc[i] + bias[i & (D1 - 1)]);
}

// ---------------------------------------------------------------------------
__global__ void k_init_pool(float* __restrict__ sums, float* __restrict__ cnt, int NG) {
  int i = blockIdx.x * blockDim.x + threadIdx.x;
  if (i < NG * D1) sums[i] = 0.f;
  if (i < NG) cnt[i] = 0.f;
}

__global__ __launch_bounds__(256) void k_pool(const float* __restrict__ h,
                                              const long long* __restrict__ batch,
                                              float* __restrict__ sums,
                                              float* __restrict__ cnt) {
  int n = blockIdx.x, j = threadIdx.x;
  int g = (int)batch[n];
  atomicAdd(&sums[g * D1 + j], h[n * D1 + j]);
  if (j == 0) atomicAdd(&cnt[g], 1.f);
}

__global__ void k_pool_div(const float* __restrict__ sums, const float* __restrict__ cnt,
                           float* __restrict__ pooled, int NG) {
  int i = blockIdx.x * blockDim.x + threadIdx.x;
  if (i >= NG * D1) return;
  pooled[i] = sums[i] / fmaxf(cnt[i / D1], 1.f);
}

// ---------------------------------------------------------------------------
// KAN layer 1: [NG,256] -> [NG,32]; one block per graph; basis staged in LDS.
// ---------------------------------------------------------------------------
__global__ __launch_bounds__(256) void k_kan1(const float* __restrict__ pooled,
                                              const float* __restrict__ base_w,
                                              const float* __restrict__ spline_w,
                                              const float* __restrict__ scaler,
                                              float* __restrict__ z) {
  __shared__ float s_spl[D1][8];
  __shared__ float s_act[D1];
  __shared__ float s_red[8][32];
  int n = blockIdx.x, t = threadIdx.x;
  float x = pooled[n * D1 + t];
  float bs[8];
  bspline8(x, bs);
#pragma unroll
  for (int p = 0; p < 8; ++p) s_spl[t][p] = bs[p];
  s_act[t] = silu(x);
  __syncthreads();
  int o = t & 31, chunk = t >> 5;
  float acc = 0.f;
  for (int ii = 0; ii < 32; ++ii) {
    int i = chunk * 32 + ii;
    const float* sw = spline_w + ((size_t)o * D1 + i) * 8;
    float sp = 0.f;
#pragma unroll
    for (int p = 0; p < 8; ++p) sp += s_spl[i][p] * sw[p];
    acc += s_act[i] * base_w[o * D1 + i] + scaler[o * D1 + i] * sp;
  }
  s_red[chunk][o] = acc;
  __syncthreads();
  if (t < 32) {
    float r = 0.f;
#pragma unroll
    for (int c = 0; c < 8; ++c) r += s_red[c][t];
    z[n * 32 + t] = r;
  }
}

// KAN layer 2: [NG,32] -> [NG,1]; one thread per graph.
__global__ void k_kan2(const float* __restrict__ z, const float* __restrict__ base_w,
                       const float* __restrict__ spline_w,
                       const float* __restrict__ scaler, float* __restrict__ out, int NG) {
  int n = blockIdx.x * blockDim.x + threadIdx.x;
  if (n >= NG) return;
  float acc = 0.f;
  for (int i = 0; i < 32; ++i) {
    float x = z[n * 32 + i];
    float bs[8];
    bspline8(x, bs);
    float sp = 0.f;
#pragma unroll
    for (int p = 0; p < 8; ++p) sp += bs[p] * spline_w[i * 8 + p];
    acc += silu(x) * base_w[i] + scaler[i] * sp;
  }
  out[n] = acc;
}

// ---------------------------------------------------------------------------
extern "C" void kernel_launch(void* const* d_in, const int* in_sizes, int n_in,
                              void* d_out, int out_size, void* d_ws, size_t ws_size,
                              hipStream_t stream) {
  const float*     x      = (const float*)d_in[0];
  const long long* ei     = (const long long*)d_in[1];
  const long long* batch  = (const long long*)d_in[2];
  const float*     W1     = (const float*)d_in[3];
  const float*     asrc1  = (const float*)d_in[4];
  const float*     adst1  = (const float*)d_in[5];
  const float*     b1     = (const float*)d_in[6];
  const float*     W2     = (const float*)d_in[7];
  const float*     asrc2  = (const float*)d_in[8];
  const float*     adst2  = (const float*)d_in[9];
  const float*     b2     = (const float*)d_in[10];
  const float*     basew1 = (const float*)d_in[11];
  const float*     splw1  = (const float*)d_in[12];
  const float*     scal1  = (const float*)d_in[13];
  const float*     basew2 = (const float*)d_in[14];
  const float*     splw2  = (const float*)d_in[15];
  const float*     scal2  = (const float*)d_in[16];
  float* out = (float*)d_out;

  const int N  = in_sizes[0] / 6;   // 50000
  const int E  = in_sizes[1] / 2;   // 800000
  const int NG = out_size;          // 1024

  // ---- workspace carving (~156 MB, all (re)initialized every call) --------
  float* ws = (float*)d_ws;
  size_t o = 0;
  float*    xw    = ws + o; o += (size_t)N * D1;
  float*    accb  = ws + o; o += (size_t)N * D1;
  float*    hbuf  = ws + o; o += (size_t)N * D1;
  float*    as    = ws + o; o += (size_t)N * HEADS;
  float*    ad    = ws + o; o += (size_t)N * HEADS;
  unsigned* mkey  = (unsigned*)(ws + o); o += (size_t)N * HEADS;
  float*    den   = ws + o; o += (size_t)N * HEADS;
  float*    sums  = ws + o; o += (size_t)NG * D1;
  float*    cnt   = ws + o; o += (size_t)NG;
  float*    pooled= ws + o; o += (size_t)NG * D1;
  float*    zbuf  = ws + o; o += (size_t)NG * 32;
  (void)ws_size; (void)n_in;

  const int nodeElems   = N * D1;
  const int nodeBlocks  = (nodeElems + 255) / 256;
  const int edgeBlocks  = (E + N + 255) / 256;
  const int scatterGrid = 8192;
  const int tilesM      = (N + 15) / 16;             // 3125
  const dim3 wmmaGrid((tilesM + 7) / 8, 4);          // 128-row strips x 4 col panels

  // =================== GAT layer 1 ===================
  k_gemm_small<<<N, 256, 0, stream>>>(x, W1, xw);
  k_alpha<<<N, 256, 0, stream>>>(xw, asrc1, adst1, as, ad);
  k_init_nodes<<<nodeBlocks, 256, 0, stream>>>(accb, mkey, den, N);
  k_edge_max<<<edgeBlocks, 256, 0, stream>>>(ei, E, N, as, ad, mkey);
  k_edge_exp<<<edgeBlocks, 256, 0, stream>>>(ei, E, N, as, ad, mkey, den);
  k_edge_scatter<<<scatterGrid, 256, 0, stream>>>(ei, E, N, as, ad, mkey, den, xw, accb);
  k_bias_elu<<<nodeBlocks, 256, 0, stream>>>(accb, b1, hbuf, nodeElems);

  // =================== GAT layer 2 (WMMA GEMM, LDS-staged B) ===================
  k_gemm_wmma<<<wmmaGrid, 256, 0, stream>>>(hbuf, W2, xw, N);
  k_alpha<<<N, 256, 0, stream>>>(xw, asrc2, adst2, as, ad);
  k_init_nodes<<<nodeBlocks, 256, 0, stream>>>(accb, mkey, den, N);
  k_edge_max<<<edgeBlocks, 256, 0, stream>>>(ei, E, N, as, ad, mkey);
  k_edge_exp<<<edgeBlocks, 256, 0, stream>>>(ei, E, N, as, ad, mkey, den);
  k_edge_scatter<<<scatterGrid, 256, 0, stream>>>(ei, E, N, as, ad, mkey, den, xw, accb);
  k_bias_elu<<<nodeBlocks, 256, 0, stream>>>(accb, b2, hbuf, nodeElems);

  // =================== mean pool + KAN head ===================
  k_init_pool<<<(NG * D1 + 255) / 256, 256, 0, stream>>>(sums, cnt, NG);
  k_pool<<<N, 256, 0, stream>>>(hbuf, batch, sums, cnt);
  k_pool_div<<<(NG * D1 + 255) / 256, 256, 0, stream>>>(sums, cnt, pooled, NG);
  k_kan1<<<NG, 256, 0, stream>>>(pooled, basew1, splw1, scal1, zbuf);
  k_kan2<<<(NG + 255) / 256, 256, 0, stream>>>(zbuf, basew2, splw2, scal2, out, NG);
}